// ShellProvider_17884243820650
// MI455X (gfx1250) — compile-verified
//
#include <hip/hip_runtime.h>

typedef __attribute__((ext_vector_type(2))) float v2f;
typedef __attribute__((ext_vector_type(4))) float v4f;
typedef __attribute__((ext_vector_type(8))) float v8f;

#define BATCH 16
#define NATOM 1024

// One wave32 per 16x16 (i,j) tile of one batch system.
// D = A x B via V_WMMA_F32_16X16X4_F32 gives the Gram matrix G[m][n] = p_i[m].p_j[n]
// (K=0..2 real, K=3 zero pad).  sq = (|pi|^2 + |pj|^2 - 2G) * mask, dist = sqrt(sq).
__global__ __launch_bounds__(256) void shell_provider_kernel(
    const float* __restrict__ pos,   // (B, A, 3)
    const int*   __restrict__ mask,  // (B, A, A)
    float* __restrict__ dist,        // (B, A, A)
    float* __restrict__ dvec)        // (B, A, A, 3)
{
    const int lane = threadIdx.x & 31;
    const int wave = threadIdx.x >> 5;
    const int tileId = blockIdx.x * 8 + wave;     // B * 64 * 64 = 65536 tiles
    const int tj = tileId & 63;
    const int ti = (tileId >> 6) & 63;
    const int b  = tileId >> 12;

    const int i0 = ti * 16;
    const int j0 = tj * 16;
    const int idx = lane & 15;                    // row (A) / col (B) owned by this lane
    const bool hi = lane >= 16;

    const float* pb = pos + (size_t)b * (NATOM * 3);
    const float* pi = pb + (size_t)(i0 + idx) * 3;   // A-operand row M = idx
    const float* pj = pb + (size_t)(j0 + idx) * 3;   // B-operand col N = idx

    // Load full triples once; select components arithmetically (no divergent branches,
    // EXEC stays all-ones into the WMMA).
    const float pix0 = pi[0], piy0 = pi[1], piz0 = pi[2];
    const float pjx = pj[0], pjy = pj[1], pjz = pj[2];

    // 32-bit 16x4 operand layout: lanes 0-15 -> {K0,K1}, lanes 16-31 -> {K2,K3(=0)}
    v2f a, bmat;
    a.x    = hi ? piz0 : pix0;
    a.y    = hi ? 0.0f : piy0;
    bmat.x = hi ? pjz : pjx;
    bmat.y = hi ? 0.0f : pjy;

    v8f c = {};
    // (neg_a, A, neg_b, B, c_mod, C, reuse_a, reuse_b)
    v8f d = __builtin_amdgcn_wmma_f32_16x16x4_f32(
        false, a, false, bmat, (short)0, c, false, false);

    // Per-lane fixed column j = j0 + idx; lanes 16-31 own rows M=8..15 of C/D.
    const int j = j0 + idx;
    const float nj = pjx * pjx + pjy * pjy + pjz * pjz;
    const int mbase = hi ? 8 : 0;

    // Vectorized fetch of this half's 8 i-rows: 96 contiguous bytes, 16B-aligned
    // ((i0+mbase) is a multiple of 8 -> byte offset multiple of 96).
    float prow[24];
    {
        const v4f* pq = (const v4f*)(pb + (size_t)(i0 + mbase) * 3);
#pragma unroll
        for (int t = 0; t < 6; ++t) {
            v4f q = pq[t];
            prow[t * 4 + 0] = q.x; prow[t * 4 + 1] = q.y;
            prow[t * 4 + 2] = q.z; prow[t * 4 + 3] = q.w;
        }
    }

    // Constant row strides -> immediate IOFFSETs on every access.
    const size_t rowBase = ((size_t)b * NATOM + (size_t)(i0 + mbase)) * NATOM + j;
    const int*  mrow = mask + rowBase;
    float*      drow = dist + rowBase;
    float*      vrow = dvec + rowBase * 3;

#pragma unroll
    for (int v = 0; v < 8; ++v) {
        const float px = prow[v * 3 + 0];
        const float py = prow[v * 3 + 1];
        const float pz = prow[v * 3 + 2];

        const float mi =
            (__builtin_nontemporal_load(mrow + (size_t)v * NATOM) != 0) ? 1.0f : 0.0f;

        // distance vectors (streamed, never re-read) -> merges to global_store_b96 NT
        __builtin_nontemporal_store((pjx - px) * mi, vrow + (size_t)v * (3 * NATOM) + 0);
        __builtin_nontemporal_store((pjy - py) * mi, vrow + (size_t)v * (3 * NATOM) + 1);
        __builtin_nontemporal_store((pjz - pz) * mi, vrow + (size_t)v * (3 * NATOM) + 2);

        // distance from WMMA Gram matrix: |pi|^2 + |pj|^2 - 2*G   (mask in {0,1})
        const float ni = px * px + py * py + pz * pz;
        const float sq = (ni + nj - 2.0f * d[v]) * mi;
        const float r  = __builtin_amdgcn_sqrtf(sq);   // raw v_sqrt_f32, no refinement
        __builtin_nontemporal_store(sq > 0.0f ? r : 0.0f, drow + (size_t)v * NATOM);
    }
}

extern "C" void kernel_launch(void* const* d_in, const int* in_sizes, int n_in,
                              void* d_out, int out_size, void* d_ws, size_t ws_size,
                              hipStream_t stream) {
    const float* pos  = (const float*)d_in[0];                   // (16,1024,3) f32
    const int*   mask = (const int*)d_in[1];                     // (16,1024,1024) int
    float* dist = (float*)d_out;                                 // (16,1024,1024)
    float* dvec = (float*)d_out + (size_t)BATCH * NATOM * NATOM; // (16,1024,1024,3)

    const int tiles  = BATCH * (NATOM / 16) * (NATOM / 16);      // 65536
    const int blocks = tiles / 8;                                // 8 waves / 256-thread block
    shell_provider_kernel<<<blocks, 256, 0, stream>>>(pos, mask, dist, dvec);
}